// FFMCell_47665547051502
// MI455X (gfx1250) — compile-verified
//
#include <hip/hip_runtime.h>
#include <stdint.h>

#define T_DIM    4096
#define TRACE_N  64
#define CTX_N    64
#define NTHREADS 256
#define SLICE    (TRACE_N * CTX_N)   // 4096 floats per t-slice per array
#define CHUNK_F  1024                // floats per array per chunk (256 lanes x float4)
#define NCHUNK   4                   // SLICE / CHUNK_F

typedef float v4f __attribute__((ext_vector_type(4)));

// CDNA5 async global->LDS load, GVS mode:
//   copies 16 bytes from global address base + goff to the LDS address in lds_off (tracked by ASYNCcnt)
__device__ __forceinline__ void async_load_b128(uint32_t lds_off, uint32_t goff,
                                                const float* base) {
  asm volatile("global_load_async_to_lds_b128 %0, %1, %2"
               :: "v"(lds_off), "v"(goff), "s"(base)
               : "memory");
}

__global__ __launch_bounds__(NTHREADS) void ffm_cell_kernel(
    const float* __restrict__ sre, const float* __restrict__ sim_,
    const float* __restrict__ xre, const float* __restrict__ xim,
    const float* __restrict__ a, const float* __restrict__ b,
    const int* __restrict__ iv, const int* __restrict__ jv,
    float* __restrict__ out, int* __restrict__ out_cnt) {
  __shared__ float tbl_decay[TRACE_N];
  __shared__ float tbl_cos[CTX_N];
  __shared__ float tbl_sin[CTX_N];
  __shared__ v4f   stage[2][4][NTHREADS];  // [buffer][array][lane] : 32 KiB

  const int      t   = blockIdx.x;
  const uint32_t tid = threadIdx.x;
  const int      jt  = jv[t];              // uniform -> scalar load
  const float    tf  = (float)jt;

  // ---- per-block gamma factor tables: 64 exp + 64 sincos (vs 50M naive) ----
  if (tid < TRACE_N) {
    tbl_decay[tid] = __expf(-__builtin_fabsf(a[tid]) * tf);
    const float th = b[tid] * tf;
    tbl_cos[tid] = __cosf(th);
    tbl_sin[tid] = __sinf(th);
  }
  if (tid == 0) out_cnt[t] = jt + iv[t];
  __syncthreads();

  const uint32_t slice_bytes = (uint32_t)t * (SLICE * 4u) + tid * 16u;

  auto issue = [&](int k) {
    const uint32_t go = slice_bytes + (uint32_t)k * (CHUNK_F * 4u);
    const int s = k & 1;
    async_load_b128((uint32_t)(uintptr_t)&stage[s][0][tid], go, sre);
    async_load_b128((uint32_t)(uintptr_t)&stage[s][1][tid], go, sim_);
    async_load_b128((uint32_t)(uintptr_t)&stage[s][2][tid], go, xre);
    async_load_b128((uint32_t)(uintptr_t)&stage[s][3][tid], go, xim);
  };

  auto compute = [&](int k) {
    const int s = k & 1;
    const v4f vsr = stage[s][0][tid];
    const v4f vsi = stage[s][1][tid];
    const v4f vxr = stage[s][2][tid];
    const v4f vxi = stage[s][3][tid];

    const int f     = k * CHUNK_F + (int)tid * 4;  // flat float idx in slice
    const int trace = f >> 6;                      // = k*16 + tid/16
    const int ctx   = ((int)tid & 15) * 4;

    const float d = tbl_decay[trace];
    const float gr0 = d * tbl_cos[ctx + 0], gi0 = d * tbl_sin[ctx + 0];
    const float gr1 = d * tbl_cos[ctx + 1], gi1 = d * tbl_sin[ctx + 1];
    const float gr2 = d * tbl_cos[ctx + 2], gi2 = d * tbl_sin[ctx + 2];
    const float gr3 = d * tbl_cos[ctx + 3], gi3 = d * tbl_sin[ctx + 3];

    const float nr0 = __builtin_fmaf(vsr.x, gr0, __builtin_fmaf(-vsi.x, gi0, vxr.x));
    const float ni0 = __builtin_fmaf(vsr.x, gi0, __builtin_fmaf( vsi.x, gr0, vxi.x));
    const float nr1 = __builtin_fmaf(vsr.y, gr1, __builtin_fmaf(-vsi.y, gi1, vxr.y));
    const float ni1 = __builtin_fmaf(vsr.y, gi1, __builtin_fmaf( vsi.y, gr1, vxi.y));
    const float nr2 = __builtin_fmaf(vsr.z, gr2, __builtin_fmaf(-vsi.z, gi2, vxr.z));
    const float ni2 = __builtin_fmaf(vsr.z, gi2, __builtin_fmaf( vsi.z, gr2, vxi.z));
    const float nr3 = __builtin_fmaf(vsr.w, gr3, __builtin_fmaf(-vsi.w, gi3, vxr.w));
    const float ni3 = __builtin_fmaf(vsr.w, gi3, __builtin_fmaf( vsi.w, gr3, vxi.w));

    const v4f o0 = {nr0, ni0, nr1, ni1};
    const v4f o1 = {nr2, ni2, nr3, ni3};
    // out[t][trace][ctx][re/im] interleaved; 32B-aligned B128 NT stores
    const size_t ob = ((size_t)t * SLICE + (size_t)f) * 2;
    __builtin_nontemporal_store(o0, (v4f*)(out + ob));
    __builtin_nontemporal_store(o1, (v4f*)(out + ob + 4));
  };

  // ---- 2-deep async double-buffered pipeline over 4 chunks ----
  issue(0);
  issue(1);                                            // 8 async ops in flight
  asm volatile("s_wait_asynccnt 4" ::: "memory");      // chunk 0 landed (in-order)
  compute(0);
  issue(2);
  asm volatile("s_wait_asynccnt 4" ::: "memory");      // chunk 1 landed
  compute(1);
  issue(3);
  asm volatile("s_wait_asynccnt 4" ::: "memory");      // chunk 2 landed
  compute(2);
  asm volatile("s_wait_asynccnt 0" ::: "memory");      // chunk 3 landed
  compute(3);
}

extern "C" void kernel_launch(void* const* d_in, const int* in_sizes, int n_in,
                              void* d_out, int out_size, void* d_ws, size_t ws_size,
                              hipStream_t stream) {
  const float* sre  = (const float*)d_in[0];
  const float* sim_ = (const float*)d_in[1];
  const float* xre  = (const float*)d_in[2];
  const float* xim  = (const float*)d_in[3];
  const float* a    = (const float*)d_in[4];
  const float* b    = (const float*)d_in[5];
  const int*   iv   = (const int*)d_in[6];
  const int*   jv   = (const int*)d_in[7];

  float* out = (float*)d_out;
  // second tuple output (j+i, int32[T]) sits after the [T,TRACE,CTX,2] floats
  int* out_cnt = (int*)(out + (size_t)T_DIM * TRACE_N * CTX_N * 2);

  ffm_cell_kernel<<<dim3(T_DIM), dim3(NTHREADS), 0, stream>>>(
      sre, sim_, xre, xim, a, b, iv, jv, out, out_cnt);
}